// ProteinContacts_32873679683865
// MI455X (gfx1250) — compile-verified
//
#include <hip/hip_runtime.h>
#include <hip/hip_bf16.h>
#include <math.h>

// CDNA5 / gfx1250: wave32, WMMA f32 16x16x4 computes a 16x16 tile of squared
// pairwise distances via the Gram identity:
//   A row j = [x_j, y_j, z_j, |Xj|^2 (+mask penalty)]   (16x4)
//   B col i = [-2x_i, -2y_i, -2z_i, 1]                  (4x16)
//   C[j][i] = |Xi|^2                                    (16x16)
// => D[j][i] = |Xj|^2 - 2 Xj.Xi + |Xi|^2  (masked j pushed to ~1e12).
// Top-k selection runs on raw d^2 (sqrt is monotone); sqrt+EPS applied only
// to the 30 surviving values per row at output time.

typedef __attribute__((ext_vector_type(2))) float v2f;
typedef __attribute__((ext_vector_type(8))) float v8f;

#define KTOP 30
#define KPAD 32         // padded list size: 8 x float4 for ds_load_b128 scans
#define WPB  4          // waves per block
#define BLK  (WPB * 32) // 128 threads
#define FBIG 3.4e38f

// ---- prep: pack [x, y, z, |X|^2 + (1-mask)*1e12] per point into d_ws ----
__global__ void ProteinContacts_prep(const float* __restrict__ X,
                                     const float* __restrict__ mask,
                                     float4* __restrict__ x4, int N) {
    int j = blockIdx.x * blockDim.x + threadIdx.x;
    if (j < N) {
        float x = X[3 * j + 0];
        float y = X[3 * j + 1];
        float z = X[3 * j + 2];
        float s = x * x + y * y + z * z;
        s += (1.0f - mask[j]) * 1e12f;   // masked candidates sort last
        x4[j] = make_float4(x, y, z, s);
    }
}

// unrolled argmax over a 32-entry, 16B-aligned LDS list (8x ds_load_b128)
__device__ __forceinline__ void scan_max32(const float* __restrict__ lst,
                                           float& w0, int& p0) {
    const float4* v4 = (const float4*)lst;
    float bw = -FBIG; int bp = 0;
#pragma unroll
    for (int c = 0; c < 8; ++c) {
        float4 f = v4[c];
        if (f.x > bw) { bw = f.x; bp = 4 * c + 0; }
        if (f.y > bw) { bw = f.y; bp = 4 * c + 1; }
        if (f.z > bw) { bw = f.z; bp = 4 * c + 2; }
        if (f.w > bw) { bw = f.w; bp = 4 * c + 3; }
    }
    w0 = bw; p0 = bp;
}

// unrolled argmin over a 32-entry, 16B-aligned LDS list
__device__ __forceinline__ void scan_min32(const float* __restrict__ lst,
                                           float& w0, int& p0) {
    const float4* v4 = (const float4*)lst;
    float bw = FBIG; int bp = 0;
#pragma unroll
    for (int c = 0; c < 8; ++c) {
        float4 f = v4[c];
        if (f.x < bw) { bw = f.x; bp = 4 * c + 0; }
        if (f.y < bw) { bw = f.y; bp = 4 * c + 1; }
        if (f.z < bw) { bw = f.z; bp = 4 * c + 2; }
        if (f.w < bw) { bw = f.w; bp = 4 * c + 3; }
    }
    w0 = bw; p0 = bp;
}

// ---- main: per-wave 16 query rows, WMMA over N/16 candidate tiles ----
__global__ void __launch_bounds__(BLK)
ProteinContacts_topk(const float4* __restrict__ x4,
                     const float* __restrict__ mask,
                     float* __restrict__ out, int N) {
    // per-lane top-K candidate lists (d^2), padded to 32 for b128 scans
    __shared__ __align__(16) float sD[WPB][32][KPAD];
    __shared__ __align__(16) int   sI[WPB][32][KPAD];

    const int lane = threadIdx.x & 31;
    const int wib  = threadIdx.x >> 5;                 // wave in block
    const int gw   = blockIdx.x * WPB + wib;           // global wave id
    const int ibase = gw * 16;                         // 16 query rows / wave
    const int icol = lane & 15;                        // query column N=lane%16
    const int half = lane >> 4;                        // 0: K0/K1+rows0-7, 1: K2/K3+rows8-15

    // ---- loop-invariant B and C (query side) ----
    float4 q = x4[ibase + icol];
    v2f B;
    // A-mirrored half-split: VGPR0 = {K0 | K2}, VGPR1 = {K1 | K3}
    B.x = half ? (-2.0f * q.z) : (-2.0f * q.x);
    B.y = half ? 1.0f          : (-2.0f * q.y);
    const float si = q.w;                              // |Xi|^2 (+penalty if row masked)
    v8f C;
#pragma unroll
    for (int v = 0; v < 8; ++v) C[v] = si;

    float* myD = sD[wib][lane];
    int*   myI = sI[wib][lane];

    // fill: real slots = +FBIG (always evictable), pad slots = -FBIG (never argmax)
#pragma unroll 1
    for (int t = 0; t < KTOP; ++t) { myD[t] = FBIG; myI[t] = 0; }
    myD[30] = -FBIG; myD[31] = -FBIG; myI[30] = 0; myI[31] = 0;
    float worst = FBIG;
    int worstPos = 0;

    for (int jb = 0; jb < N; jb += 16) {
        // candidate tile: each lane loads full float4 of j = jb + lane%16
        float4 cj = x4[jb + icol];
        if (jb + 64 < N)
            __builtin_prefetch((const void*)(x4 + jb + 64), 0, 3); // near-scope prefetch
        v2f A;
        A.x = half ? cj.z : cj.x;
        A.y = half ? cj.w : cj.y;

        // d2 tile: lane holds candidates M = v + 8*half, query column N = icol
        v8f D = __builtin_amdgcn_wmma_f32_16x16x4_f32(
            /*neg_a=*/false, A, /*neg_b=*/false, B,
            /*c_mod=*/(short)0, C, /*reuse_a=*/false, /*reuse_b=*/false);

#pragma unroll
        for (int v = 0; v < 8; ++v) {
            float d2 = D[v];                           // selection on raw d^2
            if (d2 < worst) {
                int j = jb + v + half * 8;
                myD[worstPos] = d2; myI[worstPos] = j;
                scan_max32(myD, worst, worstPos);      // 8x ds_load_b128, one wait
            }
        }
    }

    // flip pads to +FBIG so the min-select merge never picks them
    myD[30] = FBIG; myD[31] = FBIG;
    __syncthreads();

    // ---- merge lanes (l, l+16) for each query row; emit ascending top-30 ----
    if (half == 0) {
        const int i  = ibase + icol;
        const float mi = mask[i];
        float* dA = sD[wib][lane];
        int*   iA = sI[wib][lane];
        float* dB = sD[wib][lane + 16];
        int*   iB = sI[wib][lane + 16];
        const long long NK = (long long)N * KTOP;

        for (int r = 0; r < KTOP; ++r) {
            float bA, bB; int pA, pB;
            scan_min32(dA, bA, pA);
            scan_min32(dB, bB, pB);

            float d2; int j;
            if (bA <= bB) { d2 = bA; j = iA[pA]; dA[pA] = FBIG; }
            else          { d2 = bB; j = iB[pB]; dB[pB] = FBIG; }

            // reference: D = sqrt(sum(dx*dx) + EPS)
            float d = __builtin_amdgcn_sqrtf(__builtin_fmaxf(d2, 0.0f) + 1e-6f);

            out[(long long)i * KTOP + r]            = d;           // D_neighbors
            out[NK + (long long)i * KTOP + r]       = (float)j;    // E_idx (numeric)
            out[2 * NK + (long long)i * KTOP + r]   = mi * mask[j];// mask_neighbors
        }
    }
}

extern "C" void kernel_launch(void* const* d_in, const int* in_sizes, int n_in,
                              void* d_out, int out_size, void* d_ws, size_t ws_size,
                              hipStream_t stream) {
    const float* X    = (const float*)d_in[0];   // [1, N, 3]
    const float* mask = (const float*)d_in[1];   // [1, N]
    const int N = in_sizes[0] / 3;               // 8192

    float4* x4 = (float4*)d_ws;                  // N * 16 bytes of scratch

    ProteinContacts_prep<<<(N + 255) / 256, 256, 0, stream>>>(X, mask, x4, N);

    const int nWaves = N / 16;                   // 512 waves, 16 query rows each
    const int blocks = (nWaves + WPB - 1) / WPB; // 128 blocks of 128 threads
    ProteinContacts_topk<<<blocks, BLK, 0, stream>>>(x4, mask, (float*)d_out, N);
}